// GeneralConvMultiAttention_58385785422518
// MI455X (gfx1250) — compile-verified
//
#include <hip/hip_runtime.h>
#include <stdint.h>

#define D 64

typedef float v2f __attribute__((ext_vector_type(2)));
typedef float v8f __attribute__((ext_vector_type(8)));

// Order-preserving float <-> uint mapping (monotone): enables atomicMin/Max on floats.
__device__ __forceinline__ unsigned fordU(float f) {
    unsigned b = __float_as_uint(f);
    return (b & 0x80000000u) ? ~b : (b | 0x80000000u);
}
__device__ __forceinline__ float fordF(unsigned u) {
    unsigned b = (u & 0x80000000u) ? (u & 0x7FFFFFFFu) : ~u;
    return __uint_as_float(b);
}

// ---------------- init: agg=encoded(+inf-ish max), segmax=encoded min, denom=0 ------
__global__ void k_init(unsigned* __restrict__ agg_u, unsigned* __restrict__ segmax_u,
                       float* __restrict__ denom, int N) {
    int gid = blockIdx.x * blockDim.x + threadIdx.x;
    if (gid < N * D) agg_u[gid] = 0xFFFFFFFFu;
    if (gid < N) { segmax_u[gid] = 0u; denom[gid] = 0.0f; }
}

// ---------------- GEMM: y = x @ W_msg + b_msg + b_edge  via v_wmma_f32_16x16x4_f32 --
// Each wave: one 16-row tile x all 64 columns (4 N-tiles of 16). 64 WMMAs per wave.
// W is staged in LDS K-pair-interleaved so each lane's B fragment is one b64 load:
//   sWp[(k>>1)*128 + n*2 + (k&1)]  ->  lane reads contiguous {W[k0][n], W[k0+1][n]}.
__global__ __launch_bounds__(256) void k_gemm(const float* __restrict__ x,
                                              const float* __restrict__ Wm,
                                              const float* __restrict__ bm,
                                              const float* __restrict__ be,
                                              float* __restrict__ y, int N) {
    __shared__ float sWp[32 * 128];    // 16 KB, packed K-pairs
    __shared__ float sX[8 * 16 * D];   // 32 KB (8 waves x 16x64 tile)
    const int tid  = threadIdx.x;
    const int wave = tid >> 5;
    const int lane = tid & 31;
    const int half = lane >> 4;   // 0: K pair {0,1}; 1: K pair {2,3} (16x4 f32 A layout)
    const int lm   = lane & 15;

    for (int i = tid; i < D * D; i += 256) {
        int k = i >> 6, n = i & 63;
        sWp[(k >> 1) * 128 + n * 2 + (k & 1)] = Wm[i];
    }

    const int NT   = (N + 15) >> 4;
    const int tile = blockIdx.x * 8 + wave;
    const int row0 = tile * 16;
    float* sXw = sX + wave * 16 * D;
    if (tile < NT) {
        const int base = row0 * D, lim = N * D;
        for (int i = lane; i < 16 * D; i += 32) {
            int g = base + i;
            sXw[i] = (g < lim) ? x[g] : 0.0f;
        }
    }
    __syncthreads();
    if (tile >= NT) return;

    v8f z = {0.f, 0.f, 0.f, 0.f, 0.f, 0.f, 0.f, 0.f};
    v8f acc[4] = {z, z, z, z};

    for (int kk = 0; kk < 16; ++kk) {           // K = 64 in steps of 4
        const int k0 = kk * 4 + half * 2;       // this lane's K pair base
        const v2f a = *(const v2f*)&sXw[lm * D + k0];
        const float* bRow = &sWp[(kk * 2 + half) * 128];
#pragma unroll
        for (int t = 0; t < 4; ++t) {           // 4 N-tiles of 16 columns
            const v2f b = *(const v2f*)&bRow[(t * 16 + lm) * 2];
            acc[t] = __builtin_amdgcn_wmma_f32_16x16x4_f32(
                false, a, false, b, (short)0, acc[t], false, false);
        }
    }

    // C/D layout: VGPR v -> M = v + 8*half, N = lm (+16*t)
#pragma unroll
    for (int t = 0; t < 4; ++t) {
        const int col = t * 16 + lm;
        const float bias = bm[col] + be[col];
        const float* ap = (const float*)&acc[t];
#pragma unroll
        for (int v = 0; v < 8; ++v) {
            const int row = row0 + half * 8 + v;
            if (row < N) y[row * D + col] = ap[v] + bias;
        }
    }
}

// ---------------- cE = dot(W_edge[0], att)  (single wave) ---------------------------
__global__ void k_ce(const float* __restrict__ We, const float* __restrict__ att,
                     float* __restrict__ cE) {
    int lane = threadIdx.x;
    float p = We[2 * lane] * att[2 * lane] + We[2 * lane + 1] * att[2 * lane + 1];
    for (int off = 16; off > 0; off >>= 1) p += __shfl_xor(p, off, 32);
    if (lane == 0) cE[0] = p;
}

// ---------------- ydot[n] = dot(y[n], att)  (one wave32 per node, float2 per lane) --
__global__ __launch_bounds__(256) void k_ydot(const float* __restrict__ y,
                                              const float* __restrict__ att,
                                              float* __restrict__ ydot, int N) {
    int wave = threadIdx.x >> 5, lane = threadIdx.x & 31;
    int node = blockIdx.x * 8 + wave;
    if (node >= N) return;
    const float2* yp = (const float2*)(y + node * D);
    const float2* ap = (const float2*)att;
    float2 v = yp[lane], a2 = ap[lane];
    float p = v.x * a2.x + v.y * a2.y;
    for (int off = 16; off > 0; off >>= 1) p += __shfl_xor(p, off, 32);
    if (lane == 0) ydot[node] = p;
}

// ---------------- pass1: logits + segment max over dst ------------------------------
__global__ void k_pass1(const int* __restrict__ src, const int* __restrict__ dst,
                        const float* __restrict__ ea, const float* __restrict__ ydot,
                        const float* __restrict__ cE, float* __restrict__ logits,
                        unsigned* __restrict__ segmax_u, int E) {
    int e = blockIdx.x * blockDim.x + threadIdx.x;
    if (e >= E) return;
    float l = ydot[src[e]] + ea[e] * cE[0];
    l = (l > 0.0f) ? l : 0.2f * l;                 // leaky_relu(0.2)
    logits[e] = l;
    atomicMax(&segmax_u[dst[e]], fordU(l));
}

// ---------------- pass2: ex = exp(l - max); denom += ex -----------------------------
__global__ void k_pass2(const int* __restrict__ dst, float* __restrict__ exbuf,
                        const unsigned* __restrict__ segmax_u,
                        float* __restrict__ denom, int E) {
    int e = blockIdx.x * blockDim.x + threadIdx.x;
    if (e >= E) return;
    int dd = dst[e];
    float ex = __expf(exbuf[e] - fordF(segmax_u[dd]));
    exbuf[e] = ex;
    atomicAdd(&denom[dd], ex);
}

// ---------------- pass3: weighted message + segment-min scatter ---------------------
// One thread per (edge, channel): gid = e*64 + d. e is wave-uniform by construction.
__global__ __launch_bounds__(256) void k_pass3(const int* __restrict__ src,
                                               const int* __restrict__ dst,
                                               const float* __restrict__ ea,
                                               const float* __restrict__ We,
                                               const float* __restrict__ y,
                                               const float* __restrict__ exbuf,
                                               const float* __restrict__ denom,
                                               unsigned* __restrict__ agg_u, int E) {
    int gid = blockIdx.x * 256 + threadIdx.x;
    int e = gid >> 6, d = gid & 63;
    if (e >= E) return;
    int s = src[e], dd = dst[e];
    float alpha = exbuf[e] / (denom[dd] + 1e-16f);
    float w = (y[s * D + d] + ea[e] * We[d]) * alpha;
    atomicMin(&agg_u[dd * D + d], fordU(w));
}

// ---------------- finalize: out = (denom>0 ? min-agg : 0) + x -----------------------
__global__ void k_final(const float* __restrict__ x, const unsigned* __restrict__ agg_u,
                        const float* __restrict__ denom, float* __restrict__ out, int N) {
    int gid = blockIdx.x * blockDim.x + threadIdx.x;
    if (gid >= N * D) return;
    int n = gid >> 6;
    float agg = (denom[n] > 0.0f) ? fordF(agg_u[gid]) : 0.0f;
    out[gid] = x[gid] + agg;
}

extern "C" void kernel_launch(void* const* d_in, const int* in_sizes, int n_in,
                              void* d_out, int out_size, void* d_ws, size_t ws_size,
                              hipStream_t stream) {
    const float* x   = (const float*)d_in[0];
    const int*   ei  = (const int*)  d_in[1];
    const float* ea  = (const float*)d_in[2];
    const float* Wm  = (const float*)d_in[3];
    const float* bm  = (const float*)d_in[4];
    const float* We  = (const float*)d_in[5];
    const float* be  = (const float*)d_in[6];
    const float* att = (const float*)d_in[7];
    float* out = (float*)d_out;

    const int N = in_sizes[0] / D;
    const int E = in_sizes[2];
    const int* src = ei;
    const int* dst = ei + E;

    char* p = (char*)d_ws;
    float*    y        = (float*)p;    p += (size_t)N * D * 4;
    unsigned* agg_u    = (unsigned*)p; p += (size_t)N * D * 4;
    float*    ydot     = (float*)p;    p += (size_t)N * 4;
    unsigned* segmax_u = (unsigned*)p; p += (size_t)N * 4;
    float*    denom    = (float*)p;    p += (size_t)N * 4;
    float*    exbuf    = (float*)p;    p += (size_t)E * 4;
    float*    cE       = (float*)p;    p += 16;

    const int tND = N * D;
    const int NT  = (N + 15) / 16;

    k_init <<<(tND + 255) / 256, 256, 0, stream>>>(agg_u, segmax_u, denom, N);
    k_gemm <<<(NT + 7) / 8,      256, 0, stream>>>(x, Wm, bm, be, y, N);
    k_ce   <<<1,                  32, 0, stream>>>(We, att, cE);
    k_ydot <<<(N + 7) / 8,       256, 0, stream>>>(y, att, ydot, N);
    k_pass1<<<(E + 255) / 256,   256, 0, stream>>>(src, dst, ea, ydot, cE, exbuf, segmax_u, E);
    k_pass2<<<(E + 255) / 256,   256, 0, stream>>>(dst, exbuf, segmax_u, denom, E);
    k_pass3<<<(E * 64 + 255) / 256, 256, 0, stream>>>(src, dst, ea, We, y, exbuf, denom, agg_u, E);
    k_final<<<(tND + 255) / 256, 256, 0, stream>>>(x, agg_u, denom, out, N);
}